// AggregationNetwork_11355893531063
// MI455X (gfx1250) — compile-verified
//
#include <hip/hip_runtime.h>
#include <hip/hip_bf16.h>

typedef __attribute__((ext_vector_type(2))) float v2f;
typedef __attribute__((ext_vector_type(8))) float v8f;

static constexpr int NN = 100000;   // nodes
static constexpr int NE = 1600000;  // edges
static constexpr int NH = 64;       // hidden
static constexpr int NEMB = 2;      // embedding
static constexpr int NG = 64;       // graphs

// ---------------- init: deg=1 (self-loop weight), everything else 0 ----------
__global__ void k_init(float* __restrict__ deg, float* __restrict__ agg,
                       float* __restrict__ out2, float* __restrict__ misc, int miscN) {
  int tid = blockIdx.x * blockDim.x + threadIdx.x;
  int stride = gridDim.x * blockDim.x;
  for (int i = tid; i < NN; i += stride) { deg[i] = 1.0f; agg[i] = 0.0f; }
  for (int i = tid; i < 2 * NN; i += stride) out2[i] = 0.0f;
  for (int i = tid; i < miscN; i += stride) misc[i] = 0.0f;
}

// ---------------- deg[d] += ew ----------------------------------------------
__global__ void k_degree(const int* __restrict__ dst, const float* __restrict__ ew,
                         float* __restrict__ deg) {
  int e = blockIdx.x * blockDim.x + threadIdx.x;
  if (e < NE) atomicAdd(&deg[dst[e]], ew[e]);
}

// ---------------- deg -> dinv = rsqrt(deg) ----------------------------------
__global__ void k_dinv(float* __restrict__ deg) {
  int n = blockIdx.x * blockDim.x + threadIdx.x;
  if (n < NN) { float d = deg[n]; deg[n] = d > 0.0f ? rsqrtf(d) : 0.0f; }
}

// ---------------- conv1 edge pass: cache norm, scalar aggregate -------------
__global__ void k_conv1_edges(const int* __restrict__ src, const int* __restrict__ dst,
                              const float* __restrict__ ew, const float* __restrict__ dinv,
                              const float* __restrict__ x,
                              float* __restrict__ norm, float* __restrict__ agg) {
  int e = blockIdx.x * blockDim.x + threadIdx.x;
  if (e >= NE) return;
  int s = src[e], d = dst[e];
  float nm = dinv[s] * ew[e] * dinv[d];
  norm[e] = nm;
  atomicAdd(&agg[d], nm * x[s]);
}

// ---------------- self-loop + scalar BN stats (sum, sumsq of agg) -----------
__global__ void k_selfloop_stats(const float* __restrict__ dinv, const float* __restrict__ x,
                                 float* __restrict__ agg, float* __restrict__ stats) {
  __shared__ float s0[256], s1[256];
  int n = blockIdx.x * blockDim.x + threadIdx.x;
  float t = 0.0f;
  if (n < NN) {
    float di = dinv[n];
    t = agg[n] + di * di * x[n];
    agg[n] = t;
  }
  s0[threadIdx.x] = t;
  s1[threadIdx.x] = t * t;
  __syncthreads();
  for (int off = 128; off > 0; off >>= 1) {
    if (threadIdx.x < off) {
      s0[threadIdx.x] += s0[threadIdx.x + off];
      s1[threadIdx.x] += s1[threadIdx.x + off];
    }
    __syncthreads();
  }
  if (threadIdx.x == 0) { atomicAdd(&stats[0], s0[0]); atomicAdd(&stats[1], s1[0]); }
}

// ---- fold BN into per-channel affine (cf[0]=mean, cf[1+j]=A[j]) and build
// ---- a zero-padded W2^T staging array w2pad[k*16+m] so the WMMA kernel
// ---- never needs lane-predicated loads (no EXEC churn in the hot loop).
__global__ void k_coeff(const float* __restrict__ stats, const float* __restrict__ W1,
                        const float* __restrict__ gamma, const float* __restrict__ W2,
                        float* __restrict__ cf, float* __restrict__ w2pad) {
  float mean = stats[0] * (1.0f / (float)NN);
  float var  = stats[1] * (1.0f / (float)NN) - mean * mean;
  if (threadIdx.x == 0) cf[0] = mean;
  if (threadIdx.x < NH) {
    int j = threadIdx.x;
    float w = W1[j];
    cf[1 + j] = w * gamma[j] * rsqrtf(var * w * w + 1e-5f);
  }
  for (int i = threadIdx.x; i < NH * 16; i += blockDim.x) {
    int k = i >> 4, m = i & 15;
    w2pad[i] = (m < NEMB) ? W2[k * NEMB + m] : 0.0f;
  }
}

// ---------------- h2 = relu((agg-m)*A + beta) @ W2 via V_WMMA_F32_16X16X4_F32
// D[m,n] = sum_k W2[k,m] * h[node_n, k];  A-tile = padded W2^T,
// B-tile = activations of 16 nodes built in-register, 16 k-steps of 4.
// All tile-invariant operands (A fragments, BN coeffs, betas) preloaded into
// registers once per wave; wave grid-strides over tiles. EXEC stays full.
__global__ void k_h2_wmma(const float* __restrict__ agg, const float* __restrict__ cf,
                          const float* __restrict__ beta, const float* __restrict__ w2pad,
                          float* __restrict__ h2) {
  int wave = blockIdx.x * 4 + (threadIdx.x >> 5);  // 128 threads = 4 wave32
  int lane = threadIdx.x & 31;
  const int ntiles = NN / 16;                      // 6250 exact
  int totalWaves = gridDim.x * 4;
  int mrow = lane & 15;
  int koff = (lane >= 16) ? 2 : 0;
  float mean = cf[0];
  const float* A = cf + 1;

  // per-lane invariant fragments: 32 A-matrix values, 32 BN scales, 32 betas
  float aF[32], cA[32], cB[32];
#pragma unroll
  for (int s = 0; s < 16; ++s) {
    int k0 = 4 * s + koff, k1 = k0 + 1;
    aF[2 * s + 0] = w2pad[k0 * 16 + mrow];
    aF[2 * s + 1] = w2pad[k1 * 16 + mrow];
    cA[2 * s + 0] = A[k0];     cA[2 * s + 1] = A[k1];
    cB[2 * s + 0] = beta[k0];  cB[2 * s + 1] = beta[k1];
  }

  for (int tile = wave; tile < ntiles; tile += totalWaves) {  // wave-uniform
    int node = tile * 16 + mrow;
    float t = agg[node] - mean;
    v8f c = {};
#pragma unroll
    for (int s = 0; s < 16; ++s) {
      v2f bmat;
      bmat[0] = fmaxf(fmaf(t, cA[2 * s + 0], cB[2 * s + 0]), 0.0f);
      bmat[1] = fmaxf(fmaf(t, cA[2 * s + 1], cB[2 * s + 1]), 0.0f);
      v2f amat;
      amat[0] = aF[2 * s + 0];
      amat[1] = aF[2 * s + 1];
      c = __builtin_amdgcn_wmma_f32_16x16x4_f32(
          /*neg_a=*/false, amat, /*neg_b=*/false, bmat,
          /*c_mod=*/(short)0, c, /*reuse_a=*/false, /*reuse_b=*/false);
    }
    // D: VGPR0 lanes0-15 = h2[node,0], VGPR1 lanes0-15 = h2[node,1]
    if (lane < 16) {
      float2 r; r.x = c[0]; r.y = c[1];
      ((float2*)h2)[node] = r;
    }
  }
}

// ---------------- conv2 edge pass: 2-wide scatter with cached norm ----------
__global__ void k_conv2_edges(const int* __restrict__ src, const int* __restrict__ dst,
                              const float* __restrict__ norm, const float* __restrict__ h2,
                              float* __restrict__ out2) {
  int e = blockIdx.x * blockDim.x + threadIdx.x;
  if (e >= NE) return;
  int ep = e + 8192; if (ep >= NE) ep = NE - 1;
  __builtin_prefetch(&norm[ep], 0, 1);             // global_prefetch_b8
  float nm = norm[e];
  int s = src[e], d = dst[e];
  float2 hv = ((const float2*)h2)[s];
  atomicAdd(&out2[2 * d + 0], nm * hv.x);
  atomicAdd(&out2[2 * d + 1], nm * hv.y);
}

// ---------------- self-loop for conv2 + b2 + per-graph pooling (LDS bins) ---
__global__ void k_pool(const float* __restrict__ out2, const float* __restrict__ h2,
                       const float* __restrict__ dinv, const float* __restrict__ b2,
                       const int* __restrict__ batch,
                       float* __restrict__ pool, float* __restrict__ cnt) {
  __shared__ float bins[NG * NEMB];
  __shared__ float bcnt[NG];
  if (threadIdx.x < NG * NEMB) bins[threadIdx.x] = 0.0f;
  if (threadIdx.x < NG) bcnt[threadIdx.x] = 0.0f;
  __syncthreads();
  int stride = gridDim.x * blockDim.x;
  float bb0 = b2[0], bb1 = b2[1];
  for (int n = blockIdx.x * blockDim.x + threadIdx.x; n < NN; n += stride) {
    float di = dinv[n];
    float w = di * di;
    float v0 = out2[2 * n + 0] + w * h2[2 * n + 0] + bb0;
    float v1 = out2[2 * n + 1] + w * h2[2 * n + 1] + bb1;
    int g = batch[n];
    atomicAdd(&bins[2 * g + 0], v0);   // ds_add_f32 (batch sorted -> low conflict)
    atomicAdd(&bins[2 * g + 1], v1);
    atomicAdd(&bcnt[g], 1.0f);
  }
  __syncthreads();
  if (threadIdx.x < NG * NEMB) atomicAdd(&pool[threadIdx.x], bins[threadIdx.x]);
  else if (threadIdx.x < NG * NEMB + NG)
    atomicAdd(&cnt[threadIdx.x - NG * NEMB], bcnt[threadIdx.x - NG * NEMB]);
}

// ---------------- out[g,c] = pool / max(cnt,1) ------------------------------
__global__ void k_final(const float* __restrict__ pool, const float* __restrict__ cnt,
                        float* __restrict__ out) {
  int i = threadIdx.x;
  if (i < NG * NEMB) out[i] = pool[i] / fmaxf(cnt[i >> 1], 1.0f);
}

extern "C" void kernel_launch(void* const* d_in, const int* in_sizes, int n_in,
                              void* d_out, int out_size, void* d_ws, size_t ws_size,
                              hipStream_t stream) {
  (void)in_sizes; (void)n_in; (void)out_size; (void)ws_size;
  const float* x     = (const float*)d_in[0];
  const int*   ei    = (const int*)d_in[1];
  const float* ew    = (const float*)d_in[2];
  const int*   batch = (const int*)d_in[3];
  const float* W1    = (const float*)d_in[4];
  // d_in[5] = b1: cancels exactly inside BatchNorm (mean includes it)
  const float* gamma = (const float*)d_in[6];
  const float* beta  = (const float*)d_in[7];
  const float* W2    = (const float*)d_in[8];
  const float* b2    = (const float*)d_in[9];
  const int* src = ei;
  const int* dst = ei + NE;

  // workspace layout (floats): ~8.8 MB total
  float* ws    = (float*)d_ws;
  float* deg   = ws;               // [NN]   -> becomes dinv
  float* agg   = deg + NN;         // [NN]
  float* norm  = agg + NN;         // [NE]
  float* h2    = norm + NE;        // [2*NN]
  float* out2  = h2 + 2 * NN;      // [2*NN]
  float* stats = out2 + 2 * NN;    // [2]    (misc region begins here)
  float* cf    = stats + 2;        // [1+NH]
  float* pool  = cf + 1 + NH;      // [NG*NEMB]
  float* cnt   = pool + NG * NEMB; // [NG]
  float* w2pad = cnt + NG;         // [NH*16] zero-padded W2^T
  int miscN = 2 + 1 + NH + NG * NEMB + NG + NH * 16;

  k_init<<<512, 256, 0, stream>>>(deg, agg, out2, stats, miscN);
  k_degree<<<(NE + 255) / 256, 256, 0, stream>>>(dst, ew, deg);
  k_dinv<<<(NN + 255) / 256, 256, 0, stream>>>(deg);
  k_conv1_edges<<<(NE + 255) / 256, 256, 0, stream>>>(src, dst, ew, deg, x, norm, agg);
  k_selfloop_stats<<<(NN + 255) / 256, 256, 0, stream>>>(deg, x, agg, stats);
  k_coeff<<<1, 256, 0, stream>>>(stats, W1, gamma, W2, cf, w2pad);
  k_h2_wmma<<<392, 128, 0, stream>>>(agg, cf, beta, w2pad, h2);
  k_conv2_edges<<<(NE + 255) / 256, 256, 0, stream>>>(src, dst, norm, h2, out2);
  k_pool<<<256, 256, 0, stream>>>(out2, h2, deg, b2, batch, pool, cnt);
  k_final<<<1, 128, 0, stream>>>(pool, cnt, (float*)d_out);
}